// ChunkedLocalAttention_70128226009218
// MI455X (gfx1250) — compile-verified
//
#include <hip/hip_runtime.h>

#define D_MODEL 768
#define N_HEADS 12
#define HD      64
#define CHUNK   512
#define N_LM    32
#define BATCH   2
#define SEQ     8192
#define NCHUNK  (SEQ / CHUNK)   // 16
#define NKEYS   (N_LM + CHUNK)  // 544

typedef float v2f __attribute__((ext_vector_type(2)));
typedef float v4f __attribute__((ext_vector_type(4)));
typedef float v8f __attribute__((ext_vector_type(8)));

// ---------------------------------------------------------------------------
// fp32 WMMA: D(16x16) = A(16x4) * B(4x16) + C.  Wave32; even-aligned VGPRs.
// ---------------------------------------------------------------------------
static __device__ __forceinline__ v8f wmma_f32(v2f a, v2f b, v8f c) {
  return __builtin_amdgcn_wmma_f32_16x16x4_f32(
      /*neg_a=*/false, a, /*neg_b=*/false, b,
      /*c_mod=*/(short)0, c, /*reuse_a=*/false, /*reuse_b=*/false);
}

// ---------------------------------------------------------------------------
// Landmarks: lm[b][l][:] = mean over 256 consecutive tokens.
// ---------------------------------------------------------------------------
__global__ __launch_bounds__(256) void lm_kernel(const float* __restrict__ x,
                                                 float* __restrict__ lm) {
  const int b = blockIdx.x >> 5;   // BATCH*N_LM blocks
  const int l = blockIdx.x & 31;
  const int seg = SEQ / N_LM;      // 256
  const float* src = x + (size_t)(b * SEQ + l * seg) * D_MODEL;
  for (int c = threadIdx.x; c < D_MODEL; c += blockDim.x) {
    float s = 0.f;
    for (int r = 0; r < seg; ++r) s += src[(size_t)r * D_MODEL + c];
    lm[(size_t)(b * N_LM + l) * D_MODEL + c] = s * (1.0f / seg);
  }
}

#define TM  128
#define TN  64
#define TK  32
#define LDA 36   // padded LDS row stride (floats): 16B aligned, bank-spread
#define LDB 36

// ---------------------------------------------------------------------------
// Generic Y[R x 768] = X[R x 768] @ W^T + bias (used for lm K/V and Wo).
// Software-pipelined: next K-tile global loads overlap the WMMA burst.
// ---------------------------------------------------------------------------
__global__ __launch_bounds__(256) void gemm_bias_kernel(
    const float* __restrict__ X, const float* __restrict__ W,
    const float* __restrict__ bias, float* __restrict__ Y, int R) {
  __shared__ float As[TM * LDA];
  __shared__ float Bs[TN * LDB];

  const int tid  = threadIdx.x;
  const int lane = tid & 31, wid = tid >> 5;
  const int wm = wid & 3, wn = wid >> 2;       // 4x2 wave grid
  const int lr = lane & 15, half = lane >> 4;
  const int m0 = blockIdx.x * TM, n0 = blockIdx.y * TN;

  const v8f vzero = {0, 0, 0, 0, 0, 0, 0, 0};
  v8f acc[2][2];
  acc[0][0] = vzero; acc[0][1] = vzero; acc[1][0] = vzero; acc[1][1] = vzero;

  v4f xr[4], wr[2];
  auto load_tile = [&](int k0) {
#pragma unroll
    for (int i = 0; i < 4; ++i) {
      const int idx = tid + i * 256;
      const int row = idx >> 3, c4 = (idx & 7) << 2;
      v4f v = {0.f, 0.f, 0.f, 0.f};
      if (m0 + row < R)
        v = *(const v4f*)(X + (size_t)(m0 + row) * D_MODEL + k0 + c4);
      xr[i] = v;
    }
#pragma unroll
    for (int i = 0; i < 2; ++i) {
      const int idx = tid + i * 256;
      const int row = idx >> 3, c4 = (idx & 7) << 2;
      wr[i] = *(const v4f*)(W + (size_t)(n0 + row) * D_MODEL + k0 + c4);
    }
  };

  load_tile(0);
  for (int k0 = 0; k0 < D_MODEL; k0 += TK) {
#pragma unroll
    for (int i = 0; i < 4; ++i) {
      const int idx = tid + i * 256;
      *(v4f*)(As + (idx >> 3) * LDA + ((idx & 7) << 2)) = xr[i];
    }
#pragma unroll
    for (int i = 0; i < 2; ++i) {
      const int idx = tid + i * 256;
      *(v4f*)(Bs + (idx >> 3) * LDB + ((idx & 7) << 2)) = wr[i];
    }
    __syncthreads();
    if (k0 + TK < D_MODEL) load_tile(k0 + TK);  // overlap with WMMA below

#pragma unroll
    for (int kk = 0; kk < TK; kk += 4) {
      v2f a[2], b[2];
#pragma unroll
      for (int i = 0; i < 2; ++i)
        a[i] = *(const v2f*)(As + (wm * 32 + i * 16 + lr) * LDA + kk + 2 * half);
#pragma unroll
      for (int j = 0; j < 2; ++j)
        b[j] = *(const v2f*)(Bs + (wn * 32 + j * 16 + lr) * LDB + kk + 2 * half);
#pragma unroll
      for (int i = 0; i < 2; ++i)
#pragma unroll
        for (int j = 0; j < 2; ++j) acc[i][j] = wmma_f32(a[i], b[j], acc[i][j]);
    }
    __syncthreads();
  }

#pragma unroll
  for (int i = 0; i < 2; ++i)
#pragma unroll
    for (int j = 0; j < 2; ++j)
#pragma unroll
      for (int v = 0; v < 8; ++v) {
        const int r = m0 + wm * 32 + i * 16 + v + 8 * half;
        const int c = n0 + wn * 32 + j * 16 + lr;
        if (r < R) Y[(size_t)r * D_MODEL + c] = acc[i][j][v] + bias[c];
      }
}

// ---------------------------------------------------------------------------
// Fused Q/K/V projection: reads each X tile once, applies 3 weight matrices.
// Cuts x HBM traffic from 150 MB to 50 MB (the dominant saving at 23.3 TB/s).
// ---------------------------------------------------------------------------
__global__ __launch_bounds__(256) void qkv_gemm_kernel(
    const float* __restrict__ X,
    const float* __restrict__ Wq, const float* __restrict__ bq,
    const float* __restrict__ Wk, const float* __restrict__ bk,
    const float* __restrict__ Wv, const float* __restrict__ bv,
    float* __restrict__ Qo, float* __restrict__ Ko, float* __restrict__ Vo,
    int R) {
  __shared__ float As[TM * LDA];
  __shared__ float Bs[3][TN * LDB];

  const int tid  = threadIdx.x;
  const int lane = tid & 31, wid = tid >> 5;
  const int wm = wid & 3, wn = wid >> 2;
  const int lr = lane & 15, half = lane >> 4;
  const int m0 = blockIdx.x * TM, n0 = blockIdx.y * TN;

  const float* Wt[3] = {Wq, Wk, Wv};
  const float* bt[3] = {bq, bk, bv};
  float*       Yt[3] = {Qo, Ko, Vo};

  const v8f vzero = {0, 0, 0, 0, 0, 0, 0, 0};
  v8f acc[3][2][2];
#pragma unroll
  for (int w = 0; w < 3; ++w)
    acc[w][0][0] = acc[w][0][1] = acc[w][1][0] = acc[w][1][1] = vzero;

  v4f xr[4], wr[3][2];
  auto load_tile = [&](int k0) {
#pragma unroll
    for (int i = 0; i < 4; ++i) {
      const int idx = tid + i * 256;
      const int row = idx >> 3, c4 = (idx & 7) << 2;
      v4f v = {0.f, 0.f, 0.f, 0.f};
      if (m0 + row < R)
        v = *(const v4f*)(X + (size_t)(m0 + row) * D_MODEL + k0 + c4);
      xr[i] = v;
    }
#pragma unroll
    for (int w = 0; w < 3; ++w)
#pragma unroll
      for (int i = 0; i < 2; ++i) {
        const int idx = tid + i * 256;
        const int row = idx >> 3, c4 = (idx & 7) << 2;
        wr[w][i] = *(const v4f*)(Wt[w] + (size_t)(n0 + row) * D_MODEL + k0 + c4);
      }
  };

  load_tile(0);
  for (int k0 = 0; k0 < D_MODEL; k0 += TK) {
#pragma unroll
    for (int i = 0; i < 4; ++i) {
      const int idx = tid + i * 256;
      *(v4f*)(As + (idx >> 3) * LDA + ((idx & 7) << 2)) = xr[i];
    }
#pragma unroll
    for (int w = 0; w < 3; ++w)
#pragma unroll
      for (int i = 0; i < 2; ++i) {
        const int idx = tid + i * 256;
        *(v4f*)(Bs[w] + (idx >> 3) * LDB + ((idx & 7) << 2)) = wr[w][i];
      }
    __syncthreads();
    if (k0 + TK < D_MODEL) load_tile(k0 + TK);  // overlap with WMMA below

#pragma unroll
    for (int kk = 0; kk < TK; kk += 4) {
      v2f a[2];
#pragma unroll
      for (int i = 0; i < 2; ++i)
        a[i] = *(const v2f*)(As + (wm * 32 + i * 16 + lr) * LDA + kk + 2 * half);
#pragma unroll
      for (int w = 0; w < 3; ++w) {
        v2f b[2];
#pragma unroll
        for (int j = 0; j < 2; ++j)
          b[j] = *(const v2f*)(Bs[w] + (wn * 32 + j * 16 + lr) * LDB + kk + 2 * half);
#pragma unroll
        for (int i = 0; i < 2; ++i)
#pragma unroll
          for (int j = 0; j < 2; ++j)
            acc[w][i][j] = wmma_f32(a[i], b[j], acc[w][i][j]);
      }
    }
    __syncthreads();
  }

#pragma unroll
  for (int w = 0; w < 3; ++w)
#pragma unroll
    for (int i = 0; i < 2; ++i)
#pragma unroll
      for (int j = 0; j < 2; ++j)
#pragma unroll
        for (int v = 0; v < 8; ++v) {
          const int r = m0 + wm * 32 + i * 16 + v + 8 * half;
          const int c = n0 + wn * 32 + j * 16 + lr;
          if (r < R) Yt[w][(size_t)r * D_MODEL + c] = acc[w][i][j][v] + bt[w][c];
        }
}

// ---------------------------------------------------------------------------
// Attention: one block per (b, chunk, head, 128-row query strip).
// 8 waves x 16 query rows; flash-style online softmax over 34 key tiles of 16.
// K/V tile staging is software-pipelined (prefetch next tile during compute).
// ---------------------------------------------------------------------------
#define LDK 68   // K/V tile row stride (floats)
#define PLD 18   // per-wave P scratch row stride

__global__ __launch_bounds__(256) void attn_kernel(
    const float* __restrict__ Q, const float* __restrict__ Kx,
    const float* __restrict__ Klm, const float* __restrict__ Vx,
    const float* __restrict__ Vlm, float* __restrict__ AO) {
  __shared__ float Ks[16 * LDK];
  __shared__ float Vs[16 * LDK];
  __shared__ float Ps[8][16 * PLD];

  int bx = blockIdx.x;
  const int qt = bx & 3;       bx >>= 2;
  const int h  = bx % N_HEADS; bx /= N_HEADS;
  const int ck = bx % NCHUNK;
  const int b  = bx / NCHUNK;

  const int tid  = threadIdx.x;
  const int lane = tid & 31, wid = tid >> 5;
  const int lr = lane & 15, half = lane >> 4;

  const int q0   = ck * CHUNK + qt * 128 + wid * 16;  // wave's first query row
  const int hoff = h * HD;

  // Q fragments for the wave's 16 rows: A layout, K-steps of 4 over hd=64.
  v2f qf[16];
  {
    const float* qp = Q + (size_t)(b * SEQ + q0 + lr) * D_MODEL + hoff + 2 * half;
#pragma unroll
    for (int s = 0; s < 16; ++s) qf[s] = *(const v2f*)(qp + 4 * s);
  }

  const v8f vzero = {0, 0, 0, 0, 0, 0, 0, 0};
  v8f o[4] = {vzero, vzero, vzero, vzero};   // O accum: 16 rows x 64 cols
  float m_run[8], l_run[8];
#pragma unroll
  for (int v = 0; v < 8; ++v) { m_run[v] = -1e30f; l_run[v] = 0.f; }

  const float scale = 0.125f;  // hd^-0.5 = 1/8

  v4f kr[2], vr[2];
  auto load_kv = [&](int t) {
#pragma unroll
    for (int i = 0; i < 2; ++i) {
      const int idx = tid + i * 256;
      const int key = idx >> 4, c4 = (idx & 15) << 2;
      const int kidx = t * 16 + key;
      const float *kp, *vp;
      if (kidx < N_LM) {
        const size_t off = (size_t)(b * N_LM + kidx) * D_MODEL + hoff + c4;
        kp = Klm + off; vp = Vlm + off;
      } else {
        const size_t off =
            (size_t)(b * SEQ + ck * CHUNK + (kidx - N_LM)) * D_MODEL + hoff + c4;
        kp = Kx + off; vp = Vx + off;
      }
      kr[i] = *(const v4f*)kp;
      vr[i] = *(const v4f*)vp;
    }
  };

  load_kv(0);
  for (int t = 0; t < NKEYS / 16; ++t) {
#pragma unroll
    for (int i = 0; i < 2; ++i) {
      const int idx = tid + i * 256;
      const int key = idx >> 4, c4 = (idx & 15) << 2;
      *(v4f*)(Ks + key * LDK + c4) = kr[i];
      *(v4f*)(Vs + key * LDK + c4) = vr[i];
    }
    __syncthreads();
    if (t + 1 < NKEYS / 16) load_kv(t + 1);  // overlap with compute below

    // S(16x16) = Q Kt : B frag B[k][n] = K[n][k] -> contiguous in Ks row.
    v8f s = vzero;
#pragma unroll
    for (int ks = 0; ks < 16; ++ks) {
      const v2f kb = *(const v2f*)(Ks + lr * LDK + 4 * ks + 2 * half);
      s = wmma_f32(qf[ks], kb, s);
    }

    // Online softmax per row (row = v + 8*half; cols spread over 16 lanes).
    float alpha[8];
#pragma unroll
    for (int v = 0; v < 8; ++v) {
      const float e = s[v] * scale;
      float mx = e;
      mx = fmaxf(mx, __shfl_xor(mx, 1, 32));
      mx = fmaxf(mx, __shfl_xor(mx, 2, 32));
      mx = fmaxf(mx, __shfl_xor(mx, 4, 32));
      mx = fmaxf(mx, __shfl_xor(mx, 8, 32));
      const float mn = fmaxf(m_run[v], mx);
      const float a  = __expf(m_run[v] - mn);
      const float p  = __expf(e - mn);
      float rs = p;
      rs += __shfl_xor(rs, 1, 32);
      rs += __shfl_xor(rs, 2, 32);
      rs += __shfl_xor(rs, 4, 32);
      rs += __shfl_xor(rs, 8, 32);
      l_run[v] = l_run[v] * a + rs;
      m_run[v] = mn;
      alpha[v] = a;
      Ps[wid][(v + 8 * half) * PLD + lr] = p;  // C-layout -> LDS
    }
#pragma unroll
    for (int j = 0; j < 4; ++j)
#pragma unroll
      for (int v = 0; v < 8; ++v) o[j][v] *= alpha[v];

    // O += P(16x16) * V(16x64); P re-read from LDS in A layout.
#pragma unroll
    for (int ks = 0; ks < 4; ++ks) {
      const v2f pa = *(const v2f*)(&Ps[wid][lr * PLD + 4 * ks + 2 * half]);
      const int k0 = ks * 4 + 2 * half;
#pragma unroll
      for (int j = 0; j < 4; ++j) {
        v2f vb;
        vb.x = Vs[k0 * LDK + j * 16 + lr];
        vb.y = Vs[(k0 + 1) * LDK + j * 16 + lr];
        o[j] = wmma_f32(pa, vb, o[j]);
      }
    }
    __syncthreads();
  }

  // Normalize and write (B, S, H*hd) so the Wo GEMM reads it flat.
  float inv[8];
#pragma unroll
  for (int v = 0; v < 8; ++v) inv[v] = 1.0f / l_run[v];
#pragma unroll
  for (int j = 0; j < 4; ++j)
#pragma unroll
    for (int v = 0; v < 8; ++v) {
      const int r = q0 + v + 8 * half;
      AO[(size_t)(b * SEQ + r) * D_MODEL + hoff + j * 16 + lr] = o[j][v] * inv[v];
    }
}

// ---------------------------------------------------------------------------
// Host launcher
// ---------------------------------------------------------------------------
extern "C" void kernel_launch(void* const* d_in, const int* in_sizes, int n_in,
                              void* d_out, int out_size, void* d_ws, size_t ws_size,
                              hipStream_t stream) {
  (void)in_sizes; (void)n_in; (void)out_size; (void)ws_size;
  const float* x  = (const float*)d_in[0];
  const float* Wq = (const float*)d_in[1];
  const float* bq = (const float*)d_in[2];
  const float* Wk = (const float*)d_in[3];
  const float* bk = (const float*)d_in[4];
  const float* Wv = (const float*)d_in[5];
  const float* bv = (const float*)d_in[6];
  const float* Wo = (const float*)d_in[7];
  const float* bo = (const float*)d_in[8];
  float* out = (float*)d_out;
  float* ws  = (float*)d_ws;

  const size_t RTOK = (size_t)BATCH * SEQ;   // 16384
  const size_t RLM  = (size_t)BATCH * N_LM;  // 64

  float* lm  = ws;
  float* Klm = lm  + RLM * D_MODEL;
  float* Vlm = Klm + RLM * D_MODEL;
  float* Qb  = Vlm + RLM * D_MODEL;
  float* Kb  = Qb  + RTOK * D_MODEL;
  float* Vb  = Kb  + RTOK * D_MODEL;
  float* AO  = Vb  + RTOK * D_MODEL;

  lm_kernel<<<dim3(BATCH * N_LM), 256, 0, stream>>>(x, lm);

  const dim3 gX((unsigned)((RTOK + TM - 1) / TM), D_MODEL / TN);
  const dim3 gL((unsigned)((RLM + TM - 1) / TM), D_MODEL / TN);

  qkv_gemm_kernel<<<gX, 256, 0, stream>>>(x, Wq, bq, Wk, bk, Wv, bv,
                                          Qb, Kb, Vb, (int)RTOK);
  gemm_bias_kernel<<<gL, 256, 0, stream>>>(lm, Wk, bk, Klm, (int)RLM);
  gemm_bias_kernel<<<gL, 256, 0, stream>>>(lm, Wv, bv, Vlm, (int)RLM);

  attn_kernel<<<dim3(BATCH * NCHUNK * N_HEADS * (CHUNK / 128)), 256, 0, stream>>>(
      Qb, Kb, Klm, Vb, Vlm, AO);

  gemm_bias_kernel<<<gX, 256, 0, stream>>>(AO, Wo, bo, out, (int)RTOK);
}